// TensorialCPEncoder_46351287058969
// MI455X (gfx1250) — compile-verified
//
#include <hip/hip_runtime.h>

// TensorialCP encoder for MI455X (gfx1250).
// Bandwidth-bound (~280 MB traffic -> ~12us at 23.3 TB/s). Strategy:
//  - async-stage all three CP tables (196 KB) into the 320 KB WGP LDS,
//    transposing [C,R] -> [R,C] during the copy (per-lane src+dst addresses),
//  - hot loop: 6x ds_load_b128 per 4 components (2 rows x 3 axes), FMA lerp,
//  - coalesced 128-bit output stores.

#define NCOMP 32
#define RES   512
#define ROWSTRIDE 36                       // floats; 144B rows: b128-aligned, 16 bank groups
#define AXELEMS   ((RES + 1) * ROWSTRIDE)  // 513*36 = 18468 floats per axis (row 512 = zeros)
#define NTHR  1024                         // compile-time block size (matches launch)

typedef __attribute__((address_space(1))) int glb_i32;  // builtin wants AS(1) int*
typedef __attribute__((address_space(3))) int lds_i32;  // LDS side as AS(3) int*

__device__ __forceinline__ void stage_axis_async(const float* __restrict__ src_tab,
                                                 float* __restrict__ lds_axis_base,
                                                 int tid)
{
  // source is [C=32, R=512] row-major; dest is [R, ROWSTRIDE] with component fastest.
  // k = c*512 + r with r fastest across lanes -> fully coalesced 128B global reads.
#pragma unroll
  for (int it = 0; it < (NCOMP * RES) / NTHR; ++it) {   // 16 iterations
    int k = it * NTHR + tid;
    int r = k & (RES - 1);
    int c = k >> 9;
    const float* src = src_tab + k;
    float* dst = lds_axis_base + r * ROWSTRIDE + c;
    __builtin_amdgcn_global_load_async_to_lds_b32((glb_i32*)src, (lds_i32*)dst, 0, 0);
  }
}

__global__ __launch_bounds__(NTHR)
void tcp_encode_kernel(const float* __restrict__ pos,
                       const float* __restrict__ vx,
                       const float* __restrict__ vy,
                       const float* __restrict__ vz,
                       float* __restrict__ out,
                       int n)
{
  __shared__ __align__(16) float tabs[3 * AXELEMS];  // 221,616 bytes < 320 KB WGP LDS

  const int tid = threadIdx.x;

  // --- 1) zero boundary row r=512 per axis (grid_sample zero padding; the
  //     i0+1==512 edge carries weight 0, so zeros reproduce the reference). ---
  if (tid < 3 * NCOMP) {
    int a = tid >> 5;
    int c = tid & 31;
    tabs[a * AXELEMS + RES * ROWSTRIDE + c] = 0.0f;
  }

  // --- 2) async-stage tables global->LDS with transpose [C,R] -> [R,C] ---
  stage_axis_async(vx, &tabs[0 * AXELEMS], tid);
  stage_axis_async(vy, &tabs[1 * AXELEMS], tid);
  stage_axis_async(vz, &tabs[2 * AXELEMS], tid);

  asm volatile("s_wait_asynccnt 0" ::: "memory");  // ASYNCcnt not covered by barrier
  __syncthreads();

  const float* bx = &tabs[0];
  const float* by = &tabs[AXELEMS];
  const float* bz = &tabs[2 * AXELEMS];

  // --- 3) per-point interpolation + product ---
  const int stride = (int)gridDim.x * NTHR;
  for (int i = (int)blockIdx.x * NTHR + tid; i < n; i += stride) {
    float px = pos[3 * i + 0];
    float py = pos[3 * i + 1];
    float pz = pos[3 * i + 2];

    float ixf = fminf(fmaxf((px + 1.0f) * (0.5f * (RES - 1)), 0.0f), (float)(RES - 1));
    float iyf = fminf(fmaxf((py + 1.0f) * (0.5f * (RES - 1)), 0.0f), (float)(RES - 1));
    float izf = fminf(fmaxf((pz + 1.0f) * (0.5f * (RES - 1)), 0.0f), (float)(RES - 1));

    float fxi = floorf(ixf); int ix = (int)fxi; float wx = ixf - fxi;
    float fyi = floorf(iyf); int iy = (int)fyi; float wy = iyf - fyi;
    float fzi = floorf(izf); int iz = (int)fzi; float wz = izf - fzi;

    const float* x0p = bx + ix * ROWSTRIDE;  const float* x1p = x0p + ROWSTRIDE;
    const float* y0p = by + iy * ROWSTRIDE;  const float* y1p = y0p + ROWSTRIDE;
    const float* z0p = bz + iz * ROWSTRIDE;  const float* z1p = z0p + ROWSTRIDE;

    float4 o[8];
#pragma unroll
    for (int q = 0; q < 8; ++q) {
      float4 a0 = *(const float4*)(x0p + 4 * q);
      float4 a1 = *(const float4*)(x1p + 4 * q);
      float4 b0 = *(const float4*)(y0p + 4 * q);
      float4 b1 = *(const float4*)(y1p + 4 * q);
      float4 c0 = *(const float4*)(z0p + 4 * q);
      float4 c1 = *(const float4*)(z1p + 4 * q);

      float fx0 = fmaf(wx, a1.x - a0.x, a0.x);
      float fx1 = fmaf(wx, a1.y - a0.y, a0.y);
      float fx2 = fmaf(wx, a1.z - a0.z, a0.z);
      float fx3 = fmaf(wx, a1.w - a0.w, a0.w);

      float fy0 = fmaf(wy, b1.x - b0.x, b0.x);
      float fy1 = fmaf(wy, b1.y - b0.y, b0.y);
      float fy2 = fmaf(wy, b1.z - b0.z, b0.z);
      float fy3 = fmaf(wy, b1.w - b0.w, b0.w);

      float fz0 = fmaf(wz, c1.x - c0.x, c0.x);
      float fz1 = fmaf(wz, c1.y - c0.y, c0.y);
      float fz2 = fmaf(wz, c1.z - c0.z, c0.z);
      float fz3 = fmaf(wz, c1.w - c0.w, c0.w);

      o[q].x = fx0 * fy0 * fz0;
      o[q].y = fx1 * fy1 * fz1;
      o[q].z = fx2 * fy2 * fz2;
      o[q].w = fx3 * fy3 * fz3;
    }

    float4* op = (float4*)(out + (size_t)i * NCOMP);  // 128B-aligned per point
#pragma unroll
    for (int q = 0; q < 8; ++q) op[q] = o[q];
  }
}

extern "C" void kernel_launch(void* const* d_in, const int* in_sizes, int n_in,
                              void* d_out, int out_size, void* d_ws, size_t ws_size,
                              hipStream_t stream) {
  const float* pos = (const float*)d_in[0];
  const float* vx  = (const float*)d_in[1];
  const float* vy  = (const float*)d_in[2];
  const float* vz  = (const float*)d_in[3];
  float* out = (float*)d_out;
  int n = in_sizes[0] / 3;

  // 221.6 KB LDS -> one workgroup per WGP; 1024 threads = 32 wave32 waves/WGP.
  // 512 blocks: grid-stride covers 2M points (~4/thread); table staging per
  // block is ~196 KB from L2 (192 MB L2 holds everything), negligible.
  dim3 grid(512), block(NTHR);
  tcp_encode_kernel<<<grid, block, 0, stream>>>(pos, vx, vy, vz, out, n);
}